// EdgePredModel_79809082294637
// MI455X (gfx1250) — compile-verified
//
#include <hip/hip_runtime.h>
#include <hip/hip_bf16.h>

// ---------------------------------------------------------------------------
// GAT edge-predictor for MI455X (gfx1250, wave32).
//   feat = x@W via V_WMMA_F32_16X16X4_F32 (exact fp32 matrix cores)
//   edge softmax via ordered-uint atomicMax + f32 atomicAdd
//   MLP collapsed: score[e] = s_a[src] + s_b[dst] + c   (no mid-activation)
// ---------------------------------------------------------------------------

typedef float v2f __attribute__((ext_vector_type(2)));
typedef float v8f __attribute__((ext_vector_type(8)));

#define IN_F   128
#define OUT_F  32
#define HF     64   // HEADS*OUT_F
#define NEG_SLOPE 0.2f

// ---- monotone float <-> uint mapping for atomic max on floats -------------
__device__ __forceinline__ unsigned f2ord(float f) {
    unsigned u = __float_as_uint(f);
    return (u & 0x80000000u) ? ~u : (u | 0x80000000u);
}
__device__ __forceinline__ float ord2f(unsigned u) {
    return (u & 0x80000000u) ? __uint_as_float(u & 0x7FFFFFFFu)
                             : __uint_as_float(~u);
}

__device__ __forceinline__ float wave_reduce_add(float v) {
#pragma unroll
    for (int off = 16; off > 0; off >>= 1) v += __shfl_xor(v, off, 32);
    return v;
}

// ---------------------------------------------------------------------------
// K1: feat[N,64] = x[N,128] @ W[128,64] with WMMA f32 16x16x4.
// One wave -> one 16x16 tile. 4 waves/block cover the 64 output cols of the
// same 16 rows (N = 100000 is an exact multiple of 16).
// A frag (16x4 f32): lane<16 holds M=lane,K={k,k+1}; lane>=16 M=lane-16,K={k+2,k+3}
// B frag (4x16 f32): mirrored (N across lanes). C/D: 8 VGPRs, M = v + 8*(lane>=16).
// ---------------------------------------------------------------------------
__global__ __launch_bounds__(128) void k_gemm_wmma(
        const float* __restrict__ x, const float* __restrict__ W,
        float* __restrict__ feat)
{
    const int lane = threadIdx.x & 31;
    const int wave = threadIdx.x >> 5;        // 0..3 -> 16-col tile
    const int row0 = blockIdx.x * 16;
    const int col0 = wave * 16;
    const int half = lane >> 4;               // 0 or 1
    const int l16  = lane & 15;

    const float* __restrict__ xrow = x + (size_t)(row0 + l16) * IN_F + half * 2;
    const float* __restrict__ wcol = W + (size_t)(half * 2) * HF + (col0 + l16);

    v8f acc = {0.f, 0.f, 0.f, 0.f, 0.f, 0.f, 0.f, 0.f};
#pragma unroll 4
    for (int k0 = 0; k0 < IN_F; k0 += 4) {
        v2f a = *(const v2f*)(xrow + k0);               // x[row, k0+half*2 .. +1]
        v2f b;
        b.x = wcol[(size_t)k0 * HF];                    // W[k0+half*2    , col]
        b.y = wcol[(size_t)(k0 + 1) * HF];              // W[k0+half*2 + 1, col]
        acc = __builtin_amdgcn_wmma_f32_16x16x4_f32(
                /*neg_a=*/false, a, /*neg_b=*/false, b,
                /*c_mod=*/(short)0, acc, /*reuse_a=*/false, /*reuse_b=*/false);
    }

    float* out = feat + (size_t)(row0 + half * 8) * HF + (col0 + l16);
#pragma unroll
    for (int v = 0; v < 8; ++v) out[(size_t)v * HF] = acc[v];
}

// ---------------------------------------------------------------------------
// K2: per-node attention scalars. One wave per node; lane f covers feature f
// of head0 and head1. 4 wave-reductions -> el[n,2], er[n,2].
// ---------------------------------------------------------------------------
__global__ __launch_bounds__(256) void k_node_attn(
        const float* __restrict__ feat,
        const float* __restrict__ attn_l, const float* __restrict__ attn_r,
        float* __restrict__ el, float* __restrict__ er, int N)
{
    const int node = blockIdx.x * 8 + (threadIdx.x >> 5);
    const int f    = threadIdx.x & 31;
    if (node >= N) return;
    const float f0 = feat[(size_t)node * HF + f];          // head 0
    const float f1 = feat[(size_t)node * HF + OUT_F + f];  // head 1
    float l0 = wave_reduce_add(f0 * attn_l[f]);
    float l1 = wave_reduce_add(f1 * attn_l[OUT_F + f]);
    float r0 = wave_reduce_add(f0 * attn_r[f]);
    float r1 = wave_reduce_add(f1 * attn_r[OUT_F + f]);
    if (f == 0) {
        el[node * 2 + 0] = l0;  el[node * 2 + 1] = l1;
        er[node * 2 + 0] = r0;  er[node * 2 + 1] = r1;
    }
}

// ---------------------------------------------------------------------------
// K3: e = leaky_relu(el[src]+er[dst]); segment-max into m[dst] (ordered u32).
// ---------------------------------------------------------------------------
__global__ __launch_bounds__(256) void k_edge_max(
        const int* __restrict__ src, const int* __restrict__ dst,
        const float* __restrict__ el, const float* __restrict__ er,
        float* __restrict__ e, unsigned* __restrict__ m, int E)
{
    const int i = blockIdx.x * blockDim.x + threadIdx.x;
    if (i >= E) return;
    const int s = src[i], d = dst[i];
#pragma unroll
    for (int h = 0; h < 2; ++h) {
        float v = el[s * 2 + h] + er[d * 2 + h];
        v = v > 0.f ? v : NEG_SLOPE * v;
        e[(size_t)i * 2 + h] = v;
        atomicMax(&m[d * 2 + h], f2ord(v));
    }
}

// ---------------------------------------------------------------------------
// K4: ex = exp(e - m[dst]) (in place over e); denom[dst] += ex.
// ---------------------------------------------------------------------------
__global__ __launch_bounds__(256) void k_edge_exp(
        const int* __restrict__ dst, float* __restrict__ e,
        const unsigned* __restrict__ m, float* __restrict__ denom, int E)
{
    const int i = blockIdx.x * blockDim.x + threadIdx.x;
    if (i >= E) return;
    const int d = dst[i];
#pragma unroll
    for (int h = 0; h < 2; ++h) {
        float ex = __expf(e[(size_t)i * 2 + h] - ord2f(m[d * 2 + h]));
        e[(size_t)i * 2 + h] = ex;
        atomicAdd(&denom[d * 2 + h], ex);
    }
}

// ---------------------------------------------------------------------------
// K5: rst[dst,f] += feat[src,f] * (ex/denom[dst]).  64 lanes per edge ->
// coalesced gather of feat rows and coalesced f32 atomics into rst.
// ---------------------------------------------------------------------------
__global__ __launch_bounds__(256) void k_edge_aggregate(
        const int* __restrict__ src, const int* __restrict__ dst,
        const float* __restrict__ feat, const float* __restrict__ ex,
        const float* __restrict__ denom, float* __restrict__ rst, int total)
{
    const int g = blockIdx.x * blockDim.x + threadIdx.x;
    if (g >= total) return;
    const int f    = g & (HF - 1);
    const int eidx = g >> 6;
    const int s = src[eidx], d = dst[eidx];
    const int h = f >> 5;
    const float alpha = ex[(size_t)eidx * 2 + h] / denom[d * 2 + h];
    atomicAdd(&rst[(size_t)d * HF + f], feat[(size_t)s * HF + f] * alpha);
}

// ---------------------------------------------------------------------------
// K6a: collapse MLP: wab = W1@W2 (64 floats), cbias = b1@W2 + b2.
// ---------------------------------------------------------------------------
__global__ __launch_bounds__(64) void k_mlp_collapse(
        const float* __restrict__ W1, const float* __restrict__ b1,
        const float* __restrict__ W2, const float* __restrict__ b2,
        float* __restrict__ wab, float* __restrict__ cbias)
{
    const int i = threadIdx.x;   // 0..63
    float s = 0.f;
    for (int j = 0; j < OUT_F; ++j) s += W1[i * OUT_F + j] * W2[j];
    wab[i] = s;
    if (i == 0) {
        float c = b2[0];
        for (int j = 0; j < OUT_F; ++j) c += b1[j] * W2[j];
        *cbias = c;
    }
}

// ---------------------------------------------------------------------------
// K6: h = relu(mean_head(rst + bias)); s_a[n]=dot(h,wab[:32]), s_b likewise.
// One wave per node, lane = feature.
// ---------------------------------------------------------------------------
__global__ __launch_bounds__(256) void k_node_readout(
        const float* __restrict__ rst, const float* __restrict__ bias,
        const float* __restrict__ wab,
        float* __restrict__ h, float* __restrict__ sa, float* __restrict__ sb,
        int N)
{
    const int node = blockIdx.x * 8 + (threadIdx.x >> 5);
    const int f    = threadIdx.x & 31;
    if (node >= N) return;
    const float r0 = rst[(size_t)node * HF + f]          + bias[f];
    const float r1 = rst[(size_t)node * HF + OUT_F + f]  + bias[OUT_F + f];
    float hv = 0.5f * (r0 + r1);
    hv = hv > 0.f ? hv : 0.f;
    h[(size_t)node * OUT_F + f] = hv;
    float pa = wave_reduce_add(hv * wab[f]);
    float pb = wave_reduce_add(hv * wab[OUT_F + f]);
    if (f == 0) { sa[node] = pa; sb[node] = pb; }
}

// ---------------------------------------------------------------------------
// K7: score[e] = sa[src] + sb[dst] + c.
// ---------------------------------------------------------------------------
__global__ __launch_bounds__(256) void k_edge_score(
        const int* __restrict__ src, const int* __restrict__ dst,
        const float* __restrict__ sa, const float* __restrict__ sb,
        const float* __restrict__ cbias, float* __restrict__ out, int E)
{
    const int i = blockIdx.x * blockDim.x + threadIdx.x;
    if (i >= E) return;
    out[i] = sa[src[i]] + sb[dst[i]] + *cbias;
}

// ---------------------------------------------------------------------------
extern "C" void kernel_launch(void* const* d_in, const int* in_sizes, int n_in,
                              void* d_out, int out_size, void* d_ws, size_t ws_size,
                              hipStream_t stream)
{
    const float* x      = (const float*)d_in[0];
    const float* W      = (const float*)d_in[1];
    const float* attn_l = (const float*)d_in[2];
    const float* attn_r = (const float*)d_in[3];
    const float* biasg  = (const float*)d_in[4];
    const float* W1     = (const float*)d_in[5];
    const float* b1     = (const float*)d_in[6];
    const float* W2     = (const float*)d_in[7];
    const float* b2     = (const float*)d_in[8];
    const int*   src    = (const int*)d_in[9];
    const int*   dst    = (const int*)d_in[10];
    float*       out    = (float*)d_out;

    const int N = in_sizes[0] / IN_F;     // 100000
    const int E = in_sizes[9];            // 1600000

    // ---- workspace layout (floats) ----------------------------------------
    float* ws = (float*)d_ws;
    size_t o = 0;
    float*    feat  = ws + o;  o += (size_t)N * HF;
    float*    el    = ws + o;  o += (size_t)N * 2;
    float*    er    = ws + o;  o += (size_t)N * 2;
    float*    e     = ws + o;  o += (size_t)E * 2;
    unsigned* m     = (unsigned*)(ws + o);         // zeroed block starts here
    size_t zstart = o;         o += (size_t)N * 2;
    float*    denom = ws + o;  o += (size_t)N * 2;
    float*    rst   = ws + o;  o += (size_t)N * HF;
    size_t zbytes = (o - zstart) * sizeof(float);
    float*    h     = ws + o;  o += (size_t)N * OUT_F;
    float*    sa    = ws + o;  o += (size_t)N;
    float*    sb    = ws + o;  o += (size_t)N;
    float*    wab   = ws + o;  o += 64;
    float*    cbias = ws + o;  o += 1;
    (void)ws_size;

    // zero: m (ordered-uint min), denom, rst
    hipMemsetAsync((void*)m, 0, zbytes, stream);

    // collapsed MLP weights (independent of graph work)
    k_mlp_collapse<<<1, 64, 0, stream>>>(W1, b1, W2, b2, wab, cbias);

    // GAT projection GEMM (WMMA)
    k_gemm_wmma<<<(N + 15) / 16, 128, 0, stream>>>(x, W, feat);

    // per-node attention scalars
    k_node_attn<<<(N + 7) / 8, 256, 0, stream>>>(feat, attn_l, attn_r, el, er, N);

    // edge softmax
    const int eblk = (E + 255) / 256;
    k_edge_max<<<eblk, 256, 0, stream>>>(src, dst, el, er, e, m, E);
    k_edge_exp<<<eblk, 256, 0, stream>>>(dst, e, m, denom, E);

    // message aggregation: 64 lanes per edge
    const int total = E * HF;
    k_edge_aggregate<<<(total + 255) / 256, 256, 0, stream>>>(
        src, dst, feat, e, denom, rst, total);

    // node readout + collapsed per-node MLP dots
    k_node_readout<<<(N + 7) / 8, 256, 0, stream>>>(rst, biasg, wab, h, sa, sb, N);

    // per-edge scores
    k_edge_score<<<eblk, 256, 0, stream>>>(src, dst, sa, sb, cbias, out, E);
    (void)out_size;
}